// LinkPrediction_6047313953620
// MI455X (gfx1250) — compile-verified
//
#include <hip/hip_runtime.h>

typedef __attribute__((ext_vector_type(2))) float v2f;
typedef __attribute__((ext_vector_type(8))) float v8f;

namespace {
constexpr int HID       = 128;   // hidden width per node
constexpr int FEAT_COLS = 256;   // 2*HID (concat)
constexpr int N1        = 512;   // 4*HID
constexpr int M_TILE    = 64;    // edges per block
constexpr int FEAT_LD   = 260;   // padded LDS row stride (bank-conflict free)
constexpr int THREADS   = 256;   // 8 wave32 waves
}

__global__ __launch_bounds__(THREADS, 1)
void link_pred_mlp_wmma(const float* __restrict__ feat,      // [n_nodes][128]
                        const long long* __restrict__ edges, // [2][n_edges]
                        const float* __restrict__ W1,        // [256][512]
                        const float* __restrict__ b1,        // [512]
                        const float* __restrict__ W2,        // [512][2]
                        const float* __restrict__ b2,        // [2]
                        float* __restrict__ out,             // [n_edges][2]
                        int n_edges)
{
    __shared__ float sFeat[M_TILE * FEAT_LD];  // 65 KB relu(concat) tile
    __shared__ float sOut[M_TILE * 2];         // per-block output accumulator

    const int tid = threadIdx.x;
    const long long e0 = (long long)blockIdx.x * M_TILE;

    if (tid < M_TILE * 2) sOut[tid] = 0.0f;

    // ---- Gather + ReLU: 64 rows x 256 cols, float4 granularity ----
    #pragma unroll
    for (int it = 0; it < (M_TILE * FEAT_COLS / 4) / THREADS; ++it) {
        const int q   = tid + it * THREADS;
        const int row = q >> 6;        // 64 quads per edge row
        const int qi  = q & 63;
        long long e = e0 + row;
        if (e >= n_edges) e = 0;       // clamp ragged tail (store is guarded)
        const long long node = (qi < 32) ? edges[e] : edges[(long long)n_edges + e];
        const int col = (qi < 32) ? (qi * 4) : (qi * 4 - HID);
        const float4 v = *reinterpret_cast<const float4*>(feat + (size_t)node * HID + col);
        float4 r;
        r.x = fmaxf(v.x, 0.0f);
        r.y = fmaxf(v.y, 0.0f);
        r.z = fmaxf(v.z, 0.0f);
        r.w = fmaxf(v.w, 0.0f);
        *reinterpret_cast<float4*>(&sFeat[row * FEAT_LD + qi * 4]) = r;
    }
    __syncthreads();

    // ---- GEMM1 (64x256 @ 256x512) on v_wmma_f32_16x16x4_f32 ----
    const int wave  = tid >> 5;
    const int lane  = tid & 31;
    const int l16   = lane & 15;
    const int lhalf = lane >> 4;   // A/B: lanes 0-15 carry K{0,1}, 16-31 carry K{2,3}

    float ps0[4][8];  // GEMM2 partials (out column 0), per m-subtile per C-row-reg
    float ps1[4][8];  // GEMM2 partials (out column 1)
    #pragma unroll
    for (int m = 0; m < 4; ++m)
        #pragma unroll
        for (int r = 0; r < 8; ++r) { ps0[m][r] = 0.0f; ps1[m][r] = 0.0f; }

    #pragma unroll
    for (int nt = 0; nt < 4; ++nt) {
        const int n = (wave * 4 + nt) * 16 + l16;   // this lane's hidden column
        v8f acc0 = {}, acc1 = {}, acc2 = {}, acc3 = {};
        const float* wcol  = W1 + n;
        const float* fbase = &sFeat[l16 * FEAT_LD];

        for (int k = 0; k < FEAT_COLS / 4; ++k) {   // 64 K-steps of depth 4
            const int krow = k * 4 + lhalf * 2;
            v2f B;  // B 4x16: VGPR0 = row krow, VGPR1 = row krow+1 (per lane half)
            B.x = wcol[(size_t)krow * N1];
            B.y = wcol[(size_t)(krow + 1) * N1];
            v2f A;  // A 16x4: same K split, M = l16 (+ m-subtile offset)
            A.x = fbase[0 * 16 * FEAT_LD + krow];
            A.y = fbase[0 * 16 * FEAT_LD + krow + 1];
            acc0 = __builtin_amdgcn_wmma_f32_16x16x4_f32(false, A, false, B, (short)0, acc0, false, false);
            A.x = fbase[1 * 16 * FEAT_LD + krow];
            A.y = fbase[1 * 16 * FEAT_LD + krow + 1];
            acc1 = __builtin_amdgcn_wmma_f32_16x16x4_f32(false, A, false, B, (short)0, acc1, false, false);
            A.x = fbase[2 * 16 * FEAT_LD + krow];
            A.y = fbase[2 * 16 * FEAT_LD + krow + 1];
            acc2 = __builtin_amdgcn_wmma_f32_16x16x4_f32(false, A, false, B, (short)0, acc2, false, false);
            A.x = fbase[3 * 16 * FEAT_LD + krow];
            A.y = fbase[3 * 16 * FEAT_LD + krow + 1];
            acc3 = __builtin_amdgcn_wmma_f32_16x16x4_f32(false, A, false, B, (short)0, acc3, false, false);
        }

        // ---- bias + ReLU + fold GEMM2 (x W2[512x2]) into register partials ----
        const float bn  = b1[n];
        const float w20 = W2[n * 2 + 0];
        const float w21 = W2[n * 2 + 1];
        #pragma unroll
        for (int r = 0; r < 8; ++r) {
            float h;
            h = fmaxf(acc0[r] + bn, 0.0f); ps0[0][r] += h * w20; ps1[0][r] += h * w21;
            h = fmaxf(acc1[r] + bn, 0.0f); ps0[1][r] += h * w20; ps1[1][r] += h * w21;
            h = fmaxf(acc2[r] + bn, 0.0f); ps0[2][r] += h * w20; ps1[2][r] += h * w21;
            h = fmaxf(acc3[r] + bn, 0.0f); ps0[3][r] += h * w20; ps1[3][r] += h * w21;
        }
    }

    // ---- reduce hidden-column partials across the 16-lane half, then LDS atomics ----
    #pragma unroll
    for (int m = 0; m < 4; ++m) {
        #pragma unroll
        for (int r = 0; r < 8; ++r) {
            float s0 = ps0[m][r];
            float s1 = ps1[m][r];
            #pragma unroll
            for (int off = 1; off < 16; off <<= 1) {
                s0 += __shfl_xor(s0, off, 32);
                s1 += __shfl_xor(s1, off, 32);
            }
            if (l16 == 0) {
                const int row = m * 16 + lhalf * 8 + r;  // C/D layout: lanes>=16 hold M = r+8
                atomicAdd(&sOut[row * 2 + 0], s0);       // ds_add_f32
                atomicAdd(&sOut[row * 2 + 1], s1);
            }
        }
    }
    __syncthreads();

    // ---- epilogue: + b2, guarded store ----
    if (tid < M_TILE * 2) {
        const int row = tid >> 1;
        const int j   = tid & 1;
        const long long e = e0 + row;
        if (e < n_edges) out[e * 2 + j] = sOut[tid] + b2[j];
    }
}

extern "C" void kernel_launch(void* const* d_in, const int* in_sizes, int n_in,
                              void* d_out, int out_size, void* d_ws, size_t ws_size,
                              hipStream_t stream)
{
    (void)n_in; (void)out_size; (void)d_ws; (void)ws_size;
    const float*     feat  = (const float*)d_in[0];
    const long long* edges = (const long long*)d_in[1];   // int64 indices
    const float*     W1    = (const float*)d_in[2];
    const float*     b1    = (const float*)d_in[3];
    const float*     W2    = (const float*)d_in[4];
    const float*     b2    = (const float*)d_in[5];
    float* out = (float*)d_out;

    const int n_edges = in_sizes[1] / 2;                  // edges is [2][n_edges]
    const int grid = (n_edges + M_TILE - 1) / M_TILE;
    link_pred_mlp_wmma<<<grid, THREADS, 0, stream>>>(feat, edges, W1, b1, W2, b2, out, n_edges);
}